// LeafBranch_32710470927071
// MI455X (gfx1250) — compile-verified
//
#include <hip/hip_runtime.h>
#include <stdint.h>

// ---------------------------------------------------------------------------
// LeafBranch pipeline for MI455X (gfx1250, wave32)
//   k_rownorm  : per-point squared norms                      [B*N]
//   k_adj      : X·X^T via V_WMMA_F32_16X16X4_F32 with LDS tiling staged by
//                GLOBAL_LOAD_ASYNC_TO_LDS_B128 (ASYNCcnt), threshold, bit-pack
//   k_cluster  : sequential greedy label stamping (1 block)
//   k_classify : unique labels -> MT19937 downsample -> MLP -> BCE loss
// ---------------------------------------------------------------------------

typedef float v2f __attribute__((ext_vector_type(2)));
typedef float v8f __attribute__((ext_vector_type(8)));

#define BATCH   4
#define NPTS    4096
#define NF      256
#define DS      80
#define DIST2   512.1169f            // 22.63^2
#define MAXLBL  (BATCH * NPTS + 2)   // label values are < 1 + B*N
#define ADJW    (NPTS / 16)          // 256 uint16 per adjacency row

#define TM      128                  // macro-tile (rows == cols)
#define KC      32                   // K-chunk staged in LDS
#define LDSW    36                   // padded row stride: 144B (16B aligned,
                                     // 36*l16 mod 64 -> 16 distinct banks)

// ---------------------------------------------------------------------------
// Kernel 1: squared row norms. One thread per point, float4 loads.
// ---------------------------------------------------------------------------
__global__ void __launch_bounds__(256) k_rownorm(const float* __restrict__ X,
                                                 float* __restrict__ Sq) {
    int i = blockIdx.x * 256 + threadIdx.x;          // 0 .. B*N-1
    const float4* p = (const float4*)(X + (size_t)i * NF);
    float s = 0.0f;
#pragma unroll 8
    for (int k = 0; k < NF / 4; ++k) {
        float4 v = p[k];
        s += v.x * v.x + v.y * v.y + v.z * v.z + v.w * v.w;
    }
    Sq[i] = s;
}

// ---------------------------------------------------------------------------
// Kernel 2: adjacency bits via FP32 WMMA with async-LDS staged 128x128 tiles.
// Block = 256 threads = 8 waves in a 2x4 grid; each wave owns a 64x32 region
// = 4x2 sub-tiles of 16x16 (8 x v8f accumulators). Per K-chunk: stage
// A[128xKC] and B[128xKC] via GLOBAL_LOAD_ASYNC_TO_LDS_B128, wait ASYNCcnt,
// then 8 k-steps x 8 WMMAs from LDS fragments (ds_load_b64).
// A-frag (16x4 f32): lanes 0-15 M=0..15, VGPR0/1 = K = half*2+{0,1};
// B-frag mirrors with N=lane%16. C: VGPR r -> M=r (lanes 0-15) / r+8.
// ---------------------------------------------------------------------------
__global__ void __launch_bounds__(256) k_adj(const float* __restrict__ X,
                                             const float* __restrict__ Sq,
                                             uint16_t* __restrict__ adj) {
    __shared__ float As[TM * LDSW];
    __shared__ float Bs[TM * LDSW];

    int b    = blockIdx.x >> 10;          // 1024 macro-tiles per batch
    int tile = blockIdx.x & 1023;
    int tm   = tile >> 5;                 // 0..31
    int tn   = tile & 31;                 // 0..31

    int tid  = threadIdx.x;
    int lane = tid & 31;
    int wave = tid >> 5;
    int l16  = lane & 15;
    int half = lane >> 4;
    int wm   = wave >> 2;                 // 0..1 : 64-row strip
    int wn   = wave & 3;                  // 0..3 : 32-col strip

    const float* Xb = X + (size_t)b * NPTS * NF;

    v8f acc[4][2] = {};

    for (int kc = 0; kc < NF; kc += KC) {
        // ---- async stage A/B K-slices into LDS (gfx1250 ASYNCcnt path) ----
#pragma unroll
        for (int s = 0; s < 4; ++s) {
            int id   = s * 256 + tid;     // 0..1023 16B-segments per operand
            int row  = id >> 3;           // 0..127
            int segq = id & 7;            // 16B segment inside 128B row slice
            unsigned la = (unsigned)(uintptr_t)&As[row * LDSW + segq * 4];
            unsigned lb = (unsigned)(uintptr_t)&Bs[row * LDSW + segq * 4];
            const float* ga = Xb + (size_t)(tm * TM + row) * NF + kc + segq * 4;
            const float* gb = Xb + (size_t)(tn * TM + row) * NF + kc + segq * 4;
            asm volatile("global_load_async_to_lds_b128 %0, %1, off"
                         :: "v"(la), "v"(ga) : "memory");
            asm volatile("global_load_async_to_lds_b128 %0, %1, off"
                         :: "v"(lb), "v"(gb) : "memory");
        }
        asm volatile("s_wait_asynccnt 0x0" ::: "memory");
        __syncthreads();

        // ---- 8 k-steps x (4x2) WMMAs from LDS ----
#pragma unroll
        for (int kk = 0; kk < KC; kk += 4) {
            v2f av[4], bv[2];
#pragma unroll
            for (int mi = 0; mi < 4; ++mi)
                av[mi] = *(const v2f*)&As[(wm * 64 + mi * 16 + l16) * LDSW + kk + half * 2];
#pragma unroll
            for (int ni = 0; ni < 2; ++ni)
                bv[ni] = *(const v2f*)&Bs[(wn * 32 + ni * 16 + l16) * LDSW + kk + half * 2];
#pragma unroll
            for (int mi = 0; mi < 4; ++mi)
#pragma unroll
                for (int ni = 0; ni < 2; ++ni)
                    acc[mi][ni] = __builtin_amdgcn_wmma_f32_16x16x4_f32(
                        /*neg_a=*/false, av[mi], /*neg_b=*/false, bv[ni],
                        /*c_mod=*/(short)0, acc[mi][ni],
                        /*reuse_a=*/false, /*reuse_b=*/false);
        }
        __syncthreads();
    }

    // ---- threshold + wave32 ballot -> bit-packed adjacency ----
    const float* sqb = Sq + b * NPTS;
    float sqn[2];
    sqn[0] = sqb[tn * TM + wn * 32 + l16];
    sqn[1] = sqb[tn * TM + wn * 32 + 16 + l16];
#pragma unroll
    for (int mi = 0; mi < 4; ++mi) {
#pragma unroll
        for (int r = 0; r < 8; ++r) {
            float sqm = sqb[tm * TM + wm * 64 + mi * 16 + half * 8 + r];
#pragma unroll
            for (int ni = 0; ni < 2; ++ni) {
                bool pred = (sqm + sqn[ni] - 2.0f * acc[mi][ni][r]) < DIST2;
#if __has_builtin(__builtin_amdgcn_ballot_w32)
                unsigned mm = __builtin_amdgcn_ballot_w32(pred);
#else
                unsigned mm = (unsigned)__ballot(pred);
#endif
                if (lane == 0) {
                    size_t row0 = (size_t)b * NPTS + tm * TM + wm * 64 + mi * 16 + r;
                    int    cc   = tn * 8 + wn * 2 + ni;   // 16-col chunk index
                    adj[row0 * ADJW + cc]       = (uint16_t)(mm & 0xFFFFu);
                    adj[(row0 + 8) * ADJW + cc] = (uint16_t)(mm >> 16);
                }
            }
        }
    }
}

// ---------------------------------------------------------------------------
// Kernel 3: sequential greedy clustering (faithful to the reference scan).
// One block, 256 threads. Shared next-label counter persists across batches.
// ---------------------------------------------------------------------------
__global__ void __launch_bounds__(256) k_cluster(const uint16_t* __restrict__ adj,
                                                 int* __restrict__ labels) {
    __shared__ int s_next, s_new;
    int tid = threadIdx.x;
    if (tid == 0) s_next = 1;
    for (int b = 0; b < BATCH; ++b) {
        int* lb = labels + b * NPTS;
        for (int i = tid; i < NPTS; i += 256) lb[i] = 0;
        __syncthreads();
        const uint16_t* ab = adj + (size_t)b * NPTS * ADJW;
        for (int i = 0; i < NPTS; ++i) {
            if (tid == 0) {
                int cur = lb[i];
                int nw  = cur ? cur : s_next;
                if (!cur) s_next++;
                lb[i]  = nw;
                s_new  = nw;
            }
            __syncthreads();
            int nw = s_new;
            unsigned m = ab[(size_t)i * ADJW + tid];
            int base = tid * 16;
            while (m) {                    // stamp all thresholded neighbors
                int j = __ffs(m) - 1;
                lb[base + j] = nw;
                m &= m - 1;
            }
            __syncthreads();
        }
    }
}

// ---------------------------------------------------------------------------
// MT19937 (numpy legacy RandomState(0) stream), run by thread 0 only.
// ---------------------------------------------------------------------------
__device__ __forceinline__ unsigned mt_next(unsigned* mt, int& mti) {
    const unsigned MM = 397u, MATRIX = 0x9908b0dfu,
                   UPPER = 0x80000000u, LOWER = 0x7fffffffu;
    if (mti >= 624) {
        for (int k = 0; k < 624; ++k) {
            unsigned y = (mt[k] & UPPER) | (mt[(k + 1) % 624] & LOWER);
            mt[k] = mt[(k + MM) % 624] ^ (y >> 1) ^ ((y & 1u) ? MATRIX : 0u);
        }
        mti = 0;
    }
    unsigned y = mt[mti++];
    y ^= y >> 11;
    y ^= (y << 7)  & 0x9d2c5680u;
    y ^= (y << 15) & 0xefc60000u;
    y ^= y >> 18;
    return y;
}

// legacy randint: 64-bit draw (hi<<32|lo), masked rejection over [0, n)
__device__ __forceinline__ int mt_bounded(unsigned* mt, int& mti, unsigned n) {
    unsigned rng = n - 1u;
    if (rng == 0u) return 0;
    unsigned long long mask = rng;
    mask |= mask >> 1;  mask |= mask >> 2;  mask |= mask >> 4;
    mask |= mask >> 8;  mask |= mask >> 16; mask |= mask >> 32;
    unsigned long long v;
    do {
        unsigned long long hi = mt_next(mt, mti);
        unsigned long long lo = mt_next(mt, mti);
        v = ((hi << 32) | lo) & mask;
    } while (v > (unsigned long long)rng);
    return (int)v;
}

// ---------------------------------------------------------------------------
// Kernel 4: unique labels -> ordered member compaction -> MT19937 choice(80)
// -> PointNet-lite (3->64->128, relu, max-pool, 128->2) -> softmax -> BCE.
// One block, 256 threads; clusters processed strictly in np.unique order so
// the single RNG stream matches the reference's sequential consumption.
// ---------------------------------------------------------------------------
__global__ void __launch_bounds__(256) k_classify(
    const float* __restrict__ points,
    const float* __restrict__ W1, const float* __restrict__ b1,
    const float* __restrict__ W2, const float* __restrict__ b2,
    const float* __restrict__ W3, const float* __restrict__ b3,
    const int*  __restrict__ labels,
    int* __restrict__ counts, int* __restrict__ uniq, int* __restrict__ members,
    float* __restrict__ out) {

    __shared__ float W1s[3 * 64];
    __shared__ float b1s[64];
    __shared__ float W2s[64 * 128];        // 32 KB
    __shared__ float b2s[128];
    __shared__ float W3s[128 * 2];
    __shared__ float b3s[2];
    __shared__ float ptsS[DS * 3];
    __shared__ int   gI[128];              // int-punned relu max-pool (>=0)
    __shared__ int   cntbuf[257];
    __shared__ unsigned mtS[624];
    __shared__ int   s_mti, s_K;

    int tid = threadIdx.x;

    for (int i = tid; i < 3 * 64;   i += 256) W1s[i] = W1[i];
    for (int i = tid; i < 64;       i += 256) b1s[i] = b1[i];
    for (int i = tid; i < 64 * 128; i += 256) W2s[i] = W2[i];
    for (int i = tid; i < 128;      i += 256) b2s[i] = b2[i];
    for (int i = tid; i < 256;      i += 256) W3s[i] = W3[i];
    if (tid < 2) b3s[tid] = b3[tid];
    if (tid == 0) {                        // init_genrand(0)
        mtS[0] = 0u;
        for (int i = 1; i < 624; ++i)
            mtS[i] = 1812433253u * (mtS[i - 1] ^ (mtS[i - 1] >> 30)) + (unsigned)i;
        s_mti = 624;
    }
    __syncthreads();

    float total = 0.0f;                    // meaningful on tid 0 only
    for (int b = 0; b < BATCH; ++b) {
        for (int c = tid; c < MAXLBL; c += 256) counts[c] = 0;
        __syncthreads();
        const int* lb = labels + b * NPTS;
        for (int i = tid; i < NPTS; i += 256) atomicAdd(&counts[lb[i]], 1);
        __syncthreads();
        if (tid == 0) {                    // np.unique: sorted label values
            int K = 0;
            for (int c = 1; c < MAXLBL; ++c)
                if (counts[c]) uniq[K++] = c;
            s_K = K;
        }
        __syncthreads();
        int K = s_K;
        float bsum = 0.0f;

        for (int kc = 0; kc < K; ++kc) {
            int c = uniq[kc];
            // --- ordered compaction of member indices ---
            int base = tid * 16, mycnt = 0;
            for (int j = 0; j < 16; ++j) mycnt += (lb[base + j] == c);
            cntbuf[tid] = mycnt;
            __syncthreads();
            if (tid == 0) {
                int acc = 0;
                for (int t = 0; t < 256; ++t) { int v = cntbuf[t]; cntbuf[t] = acc; acc += v; }
                cntbuf[256] = acc;
            }
            __syncthreads();
            int off = cntbuf[tid];
            for (int j = 0; j < 16; ++j)
                if (lb[base + j] == c) members[off++] = base + j;
            __syncthreads();
            int Mc = cntbuf[256];

            // --- rng.choice(Mc, 80, replace=True) + gather points ---
            if (tid == 0) {
                int mti = s_mti;
                for (int j = 0; j < DS; ++j) {
                    int sel = mt_bounded(mtS, mti, (unsigned)Mc);
                    const float* pp = points + ((size_t)b * NPTS + members[sel]) * 3;
                    ptsS[j * 3 + 0] = pp[0];
                    ptsS[j * 3 + 1] = pp[1];
                    ptsS[j * 3 + 2] = pp[2];
                }
                s_mti = mti;
            }
            if (tid < 128) gI[tid] = 0;    // relu output >= 0 -> 0.0f init
            __syncthreads();

            // --- MLP: one thread per downsampled point ---
            if (tid < DS) {
                float x0 = ptsS[tid * 3 + 0];
                float x1 = ptsS[tid * 3 + 1];
                float x2 = ptsS[tid * 3 + 2];
                float h1[64];
#pragma unroll
                for (int j = 0; j < 64; ++j) {
                    float v = b1s[j] + x0 * W1s[0 * 64 + j]
                                     + x1 * W1s[1 * 64 + j]
                                     + x2 * W1s[2 * 64 + j];
                    h1[j] = v > 0.0f ? v : 0.0f;
                }
                for (int ko = 0; ko < 128; ++ko) {
                    float a = b2s[ko];
#pragma unroll 8
                    for (int j = 0; j < 64; ++j) a += h1[j] * W2s[j * 128 + ko];
                    float v = a > 0.0f ? a : 0.0f;
                    atomicMax(&gI[ko], __float_as_int(v));   // nonneg-float max
                }
            }
            __syncthreads();

            if (tid == 0) {
                float l0 = b3s[0], l1 = b3s[1];
                for (int k = 0; k < 128; ++k) {
                    float g = __int_as_float(gI[k]);
                    l0 += g * W3s[k * 2 + 0];
                    l1 += g * W3s[k * 2 + 1];
                }
                float mx = l0 > l1 ? l0 : l1;
                float e0 = __expf(l0 - mx), e1 = __expf(l1 - mx);
                float p  = e1 / (e0 + e1);
                if (isnan(p)) p = 0.0f;                     // nan_to_num
                float loss = -__logf(p);
                if (!(loss < 100.0f)) loss = 100.0f;        // clamp (inf/nan -> 100)
                bsum += loss;
            }
            __syncthreads();
        }
        if (tid == 0) total += bsum / (float)K;             // per-batch mean
        __syncthreads();
    }
    if (tid == 0) out[0] = total;
}

// ---------------------------------------------------------------------------
// Launch
// ---------------------------------------------------------------------------
extern "C" void kernel_launch(void* const* d_in, const int* in_sizes, int n_in,
                              void* d_out, int out_size, void* d_ws, size_t ws_size,
                              hipStream_t stream) {
    (void)in_sizes; (void)n_in; (void)out_size; (void)ws_size;
    const float* pf  = (const float*)d_in[0];   // point_features [4,4096,256]
    const float* pts = (const float*)d_in[1];   // points         [4,4096,3]
    const float* W1  = (const float*)d_in[2];
    const float* b1  = (const float*)d_in[3];
    const float* W2  = (const float*)d_in[4];
    const float* b2  = (const float*)d_in[5];
    const float* W3  = (const float*)d_in[6];
    const float* b3  = (const float*)d_in[7];
    float* out = (float*)d_out;

    // workspace layout (~8.6 MB)
    char* ws = (char*)d_ws;
    float*    Sq      = (float*)ws;                                  // 64 KB
    uint16_t* adj     = (uint16_t*)(ws + 64 * 1024);                 // 8 MB
    int*      labels  = (int*)(ws + 64 * 1024 + 8 * 1024 * 1024);    // 64 KB
    int*      counts  = labels + BATCH * NPTS;                       // 64 KB
    int*      uniq    = counts + ((MAXLBL + 15) & ~15);
    int*      members = uniq + NPTS;

    k_rownorm<<<(BATCH * NPTS) / 256, 256, 0, stream>>>(pf, Sq);
    // 1024 128x128 macro-tiles per batch, one per 256-thread block
    k_adj<<<BATCH * 1024, 256, 0, stream>>>(pf, Sq, adj);
    k_cluster<<<1, 256, 0, stream>>>(adj, labels);
    k_classify<<<1, 256, 0, stream>>>(pts, W1, b1, W2, b2, W3, b3,
                                      labels, counts, uniq, members, out);
}